// DrugGNN_38096359916278
// MI455X (gfx1250) — compile-verified
//
#include <hip/hip_runtime.h>
#include <math.h>

typedef __attribute__((ext_vector_type(2))) float v2f;
typedef __attribute__((ext_vector_type(8))) float v8f;

// ---------------- degree / normalization ----------------

__global__ void k_init_deg(float* __restrict__ deg, int n) {
  int i = blockIdx.x * blockDim.x + threadIdx.x;
  if (i < n) deg[i] = 1.0f;  // +1 self loop
}

__global__ void k_count_deg(const int* __restrict__ dst, float* __restrict__ deg, int e) {
  int i = blockIdx.x * blockDim.x + threadIdx.x;
  if (i < e)
    __hip_atomic_fetch_add(&deg[dst[i]], 1.0f, __ATOMIC_RELAXED, __HIP_MEMORY_SCOPE_AGENT);
}

__global__ void k_rsqrt(float* __restrict__ deg, int n) {
  int i = blockIdx.x * blockDim.x + threadIdx.x;
  if (i < n) deg[i] = rsqrtf(deg[i]);  // in place: deg -> deg^{-1/2}
}

// ---------------- fp32 WMMA GEMM: C[NxFout] = act(A[NxK]) * W[KxFout] ----------------
// One wave computes one 16x16 tile via V_WMMA_F32_16X16X4_F32 (K stepped by 4).
// A-matrix lane layout (16x4 f32): lane l -> row m0 + (l&15); lanes 0-15 hold K=k0,k0+1,
// lanes 16-31 hold K=k0+2,k0+3 (contiguous pair -> single b64 load).
// B-matrix (4x16) mirrors with N = n0 + (l&15).
// C/D: 8 VGPRs; VGPR r -> row m0 + r (lanes 0-15) / m0 + r + 8 (lanes 16-31), col n0 + (l&15).

template <bool RELU>
__global__ void k_wmma_gemm(const float* __restrict__ A, const float* __restrict__ W,
                            float* __restrict__ C, int N, int K, int Fout,
                            int totalWaves) {
  int wave = (blockIdx.x * blockDim.x + threadIdx.x) >> 5;
  if (wave >= totalWaves) return;  // wave-uniform; EXEC all-1s at the WMMA
  int lane = threadIdx.x & 31;
  int nblk = Fout >> 4;
  int m0 = (wave / nblk) << 4;
  int n0 = (wave % nblk) << 4;
  int lm = lane & 15;
  int kh = (lane >> 4) << 1;  // 0 for lanes 0-15, 2 for lanes 16-31

  v8f acc = {};
  const float* arow = A + (size_t)(m0 + lm) * K + kh;   // 8B-aligned (K,kh even)
  const float* wcol = W + (size_t)kh * Fout + (n0 + lm);

  for (int k0 = 0; k0 < K; k0 += 4) {
    v2f a = *(const v2f*)(arow + k0);   // one global_load_b64 per lane
    if (RELU) { a.x = fmaxf(a.x, 0.0f); a.y = fmaxf(a.y, 0.0f); }
    v2f b;
    b.x = wcol[(size_t)k0 * Fout];
    b.y = wcol[(size_t)(k0 + 1) * Fout];
    // (neg_a, A, neg_b, B, c_mod, C, reuse_a, reuse_b)
    acc = __builtin_amdgcn_wmma_f32_16x16x4_f32(false, a, false, b, (short)0, acc,
                                                false, false);
  }

  float* crow = C + (size_t)(m0 + ((lane >> 4) << 3)) * Fout + (n0 + lm);
#pragma unroll
  for (int r = 0; r < 8; ++r) crow[(size_t)r * Fout] = acc[r];
}

// ---------------- self-loop term + bias (also initializes agg) ----------------

__global__ void k_self_bias(const float* __restrict__ h, const float* __restrict__ dinv,
                            const float* __restrict__ bias, float* __restrict__ agg,
                            int n, int logF) {
  int idx = blockIdx.x * blockDim.x + threadIdx.x;
  if (idx >= (n << logF)) return;
  int i = idx >> logF;
  int f = idx & ((1 << logF) - 1);
  float di = dinv[i];
  agg[idx] = h[idx] * di * di + bias[f];
}

// ---------------- edge scatter: one wave per edge, one feature per lane ----------------

__global__ void k_scatter(const int* __restrict__ src, const int* __restrict__ dst,
                          const float* __restrict__ dinv, const float* __restrict__ h,
                          float* __restrict__ agg, int E, int F) {
  int wave = (blockIdx.x * blockDim.x + threadIdx.x) >> 5;
  int lane = threadIdx.x & 31;
  if (wave >= E) return;
  int s = src[wave];
  int d = dst[wave];
  float norm = dinv[s] * dinv[d];
  const float* hs = h + (size_t)s * F;
  float* ad = agg + (size_t)d * F;
  for (int f = lane; f < F; f += 32)
    __hip_atomic_fetch_add(&ad[f], hs[f] * norm, __ATOMIC_RELAXED,
                           __HIP_MEMORY_SCOPE_AGENT);
}

// ---------------- log-softmax over 32 classes: one wave per node ----------------

__global__ void k_logsoftmax32(const float* __restrict__ x, float* __restrict__ out, int N) {
  int wave = (blockIdx.x * blockDim.x + threadIdx.x) >> 5;
  int lane = threadIdx.x & 31;
  if (wave >= N) return;
  float v = x[(size_t)wave * 32 + lane];
  float m = v;
#pragma unroll
  for (int o = 16; o; o >>= 1) m = fmaxf(m, __shfl_xor(m, o, 32));
  float s = expf(v - m);
#pragma unroll
  for (int o = 16; o; o >>= 1) s += __shfl_xor(s, o, 32);
  out[(size_t)wave * 32 + lane] = v - m - logf(s);
}

// ---------------- launcher ----------------

extern "C" void kernel_launch(void* const* d_in, const int* in_sizes, int n_in,
                              void* d_out, int out_size, void* d_ws, size_t ws_size,
                              hipStream_t stream) {
  const float* x  = (const float*)d_in[0];
  const int*   ei = (const int*)d_in[1];
  const float* W1 = (const float*)d_in[2];
  const float* b1 = (const float*)d_in[3];
  const float* W2 = (const float*)d_in[4];
  const float* b2 = (const float*)d_in[5];

  const int N = in_sizes[0] / 64;   // 100000
  const int E = in_sizes[1] / 2;    // 1600000
  const int* src = ei;              // edge_index[0]
  const int* dst = ei + E;          // edge_index[1]

  float* ws   = (float*)d_ws;
  float* dinv = ws;                         // N floats (deg, then deg^{-1/2})
  float* h    = ws + N;                     // N*64 (h1; reused as h2: N*32)
  float* agg  = ws + N + (size_t)N * 64;    // N*64 (agg1; reused as agg2: N*32)
  float* out  = (float*)d_out;              // N*32

  const int BT = 256;

  // degrees -> dinv
  k_init_deg<<<(N + BT - 1) / BT, BT, 0, stream>>>(dinv, N);
  k_count_deg<<<(E + BT - 1) / BT, BT, 0, stream>>>(dst, dinv, E);
  k_rsqrt<<<(N + BT - 1) / BT, BT, 0, stream>>>(dinv, N);

  // layer 1: h1 = x @ W1 ; agg1 = b1 + h1*dinv^2 + scatter(h1[src]*norm)
  {
    int waves = (N / 16) * (64 / 16);  // 25000, exact
    k_wmma_gemm<false><<<(waves * 32 + BT - 1) / BT, BT, 0, stream>>>(x, W1, h, N, 64,
                                                                      64, waves);
    k_self_bias<<<(N * 64 + BT - 1) / BT, BT, 0, stream>>>(h, dinv, b1, agg, N, 6);
    long long t = (long long)E * 32;
    k_scatter<<<(unsigned)((t + BT - 1) / BT), BT, 0, stream>>>(src, dst, dinv, h, agg,
                                                                E, 64);
  }

  // layer 2: h2 = relu(agg1) @ W2 (relu fused into A load); agg2 then log_softmax
  {
    int waves = (N / 16) * (32 / 16);  // 12500
    k_wmma_gemm<true><<<(waves * 32 + BT - 1) / BT, BT, 0, stream>>>(agg, W2, h, N, 64,
                                                                     32, waves);
    k_self_bias<<<(N * 32 + BT - 1) / BT, BT, 0, stream>>>(h, dinv, b2, agg, N, 5);
    long long t = (long long)E * 32;
    k_scatter<<<(unsigned)((t + BT - 1) / BT), BT, 0, stream>>>(src, dst, dinv, h, agg,
                                                                E, 32);
    k_logsoftmax32<<<((long long)N * 32 + BT - 1) / BT, BT, 0, stream>>>(agg, out, N);
  }
}